// Generator_79731772883015
// MI455X (gfx1250) — compile-verified
//
#include <hip/hip_runtime.h>
#include <hip/hip_bf16.h>

// ---- problem constants (match reference) ----
#define N_VOCAB 20000
#define H       40
#define DW      128
#define ND      16
#define RW      8
#define NS      512
#define EPSG    1e-20f

// logits kernel tiling
#define STILES      (NS / 16)          // 32 sample tiles
#define NCHUNK      25                 // N split into 25 chunks
#define TILES_PER_CHUNK (N_VOCAB / 16 / NCHUNK)  // 50 tiles of 16 cols

typedef __attribute__((ext_vector_type(2))) float v2f;
typedef __attribute__((ext_vector_type(8))) float v8f;

__device__ __forceinline__ float sigmoidf_(float x) {
    return 1.0f / (1.0f + __expf(-x));
}

// combine two online-softmax partial states (max m, sum S, argmax a)
// tie-break: smaller index wins (matches jnp.argmax first-occurrence)
__device__ __forceinline__ void comb(float& m, float& S, int& a,
                                     float m2, float S2, int a2) {
    if (S2 <= 0.0f) return;
    if (S  <= 0.0f) { m = m2; S = S2; a = a2; return; }
    if (m2 > m || (m2 == m && a2 < a)) {
        S = S2 + S * __expf(m - m2);
        m = m2; a = a2;
    } else {
        S = S + S2 * __expf(m2 - m);
    }
}

// ---------------- zero the whole output (poison-safe, deterministic) --------
__global__ void zero_out_kernel(float4* __restrict__ out, long n4) {
    long i = (long)blockIdx.x * blockDim.x + threadIdx.x;
    long stride = (long)gridDim.x * blockDim.x;
    float4 z; z.x = z.y = z.z = z.w = 0.0f;
    for (; i < n4; i += stride) out[i] = z;
}

// ---------------- initial LSTM state from noise -----------------------------
__global__ __launch_bounds__(64)
void init_kernel(const float* __restrict__ z,
                 const float* __restrict__ l1w, const float* __restrict__ l1b,
                 const float* __restrict__ l2w, const float* __restrict__ l2b,
                 const float* __restrict__ l3w, const float* __restrict__ l3b,
                 float* __restrict__ h, float* __restrict__ c,
                 int* __restrict__ amax, float* __restrict__ yval) {
    int s = blockIdx.x;
    int j = threadIdx.x;
    __shared__ float zs[ND];
    __shared__ float inter[H];
    if (j < ND) zs[j] = z[s * ND + j];
    __syncthreads();
    if (j < H) {
        float acc = l1b[j];
        #pragma unroll
        for (int k = 0; k < ND; ++k) acc += zs[k] * l1w[k * H + j];
        inter[j] = tanhf(acc);
    }
    __syncthreads();
    if (j < H) {
        float a2 = l2b[j], a3 = l3b[j];
        #pragma unroll
        for (int k = 0; k < H; ++k) {
            float iv = inter[k];
            a2 += iv * l2w[k * H + j];
            a3 += iv * l3w[k * H + j];
        }
        h[s * H + j] = tanhf(a2);
        c[s * H + j] = tanhf(a3);
    }
    if (j == 0) { amax[s] = 0; yval[s] = 0.0f; }  // x0 = 0
}

// ---------------- LSTM cell (x = yval * W_down[amax]) -----------------------
__global__ __launch_bounds__(160)
void lstm_kernel(const float* __restrict__ w_ih, const float* __restrict__ w_hh,
                 const float* __restrict__ b_ih, const float* __restrict__ b_hh,
                 const float* __restrict__ W_down,
                 const int* __restrict__ amax, const float* __restrict__ yval,
                 float* __restrict__ h, float* __restrict__ c) {
    int s = blockIdx.x;
    int j = threadIdx.x;                    // 0..159 (gate column)
    __shared__ float xs[DW];
    __shared__ float hs[H];
    __shared__ float gs[4 * H];
    float yv = yval[s];
    int   am = amax[s];
    if (j < DW) xs[j] = W_down[(long)am * DW + j] * yv;
    if (j < H)  hs[j] = h[s * H + j];
    __syncthreads();
    float acc = b_ih[j] + b_hh[j];
    #pragma unroll 4
    for (int k = 0; k < DW; ++k) acc += xs[k] * w_ih[k * (4 * H) + j];
    #pragma unroll 4
    for (int k = 0; k < H; ++k)  acc += hs[k] * w_hh[k * (4 * H) + j];
    gs[j] = acc;
    __syncthreads();
    if (j < H) {
        float ig = gs[j], fg = gs[H + j], gg = gs[2 * H + j], og = gs[3 * H + j];
        float c2 = sigmoidf_(fg) * c[s * H + j] + sigmoidf_(ig) * tanhf(gg);
        float h2 = sigmoidf_(og) * tanhf(c2);
        h[s * H + j] = h2;
        c[s * H + j] = c2;
    }
}

// ---------------- logits (WMMA f32) + gumbel + online softmax partials ------
__global__ __launch_bounds__(256)
void logits_kernel(const float* __restrict__ h,
                   const float* __restrict__ W_up,
                   const float* __restrict__ b_up,
                   const float* __restrict__ u,      // gumbel_u [RW,NS,N]
                   int t,
                   float* __restrict__ pm, float* __restrict__ ps,
                   int* __restrict__ pa) {
    const int stile = blockIdx.x;           // 0..31
    const int chunk = blockIdx.y;           // 0..24
    const int tid   = threadIdx.x;
    const int wv    = tid >> 5;             // wave id 0..7
    const int lane  = tid & 31;
    const int half  = lane >> 4;            // 0: rows 0-7, 1: rows 8-15
    const int l15   = lane & 15;

    // ---- A fragments: h tile [16 x 40], K split into 10 slices of 4 ----
    // A 16x4 layout: lanes 0-15 -> M=lane,K={0,1}; lanes 16-31 -> M=lane-16,K={2,3}
    const int kb = half * 2;
    const float* hrow = h + (stile * 16 + l15) * H;
    v2f afrag[10];
    #pragma unroll
    for (int kk = 0; kk < 10; ++kk) {
        afrag[kk].x = hrow[4 * kk + kb];
        afrag[kk].y = hrow[4 * kk + kb + 1];
    }

    // per-lane online-softmax state for 8 rows (this lane-half's M values)
    float sm[8], ss[8];
    int   sa[8];
    #pragma unroll
    for (int r = 0; r < 8; ++r) { sm[r] = -__builtin_inff(); ss[r] = 0.0f; sa[r] = 0; }

    // contiguous tile split across 8 waves: waves 0-1 get 7 tiles, 2-7 get 6
    const int t0   = wv * 6 + (wv < 2 ? wv : 2);
    const int tcnt = (wv < 2) ? 7 : 6;
    const int base_n = chunk * (TILES_PER_CHUNK * 16);
    const long ubase = ((long)t * NS + (long)(stile * 16 + half * 8)) * (long)N_VOCAB;

    for (int ti = t0; ti < t0 + tcnt; ++ti) {
        const int n0 = base_n + ti * 16;
        const int n  = n0 + l15;

        v8f acc = {0.f, 0.f, 0.f, 0.f, 0.f, 0.f, 0.f, 0.f};
        #pragma unroll
        for (int kk = 0; kk < 10; ++kk) {
            // B 4x16 layout mirrors A: lanes 0-15 -> K={0,1}, lanes 16-31 -> K={2,3}
            v2f b;
            b.x = W_up[(4 * kk + kb)     * N_VOCAB + n];
            b.y = W_up[(4 * kk + kb + 1) * N_VOCAB + n];
            acc = __builtin_amdgcn_wmma_f32_16x16x4_f32(
                false, afrag[kk], false, b, (short)0, acc, false, false);
        }

        const float bup = b_up[n];
        #pragma unroll
        for (int r = 0; r < 8; ++r) {
            // C layout: VGPR r -> M = half*8 + r, N = n
            float uu = u[ubase + (long)r * N_VOCAB + n];
            float g  = -__logf(-__logf(uu + EPSG) + EPSG);
            float v  = acc[r] + bup + g;
            if (v > sm[r]) {
                ss[r] = ss[r] * __expf(sm[r] - v) + 1.0f;
                sm[r] = v; sa[r] = n;
            } else {
                ss[r] += __expf(v - sm[r]);
            }
        }
    }

    // reduce across the 16 lanes of each half (xor masks < 16 stay in-half)
    #pragma unroll
    for (int r = 0; r < 8; ++r) {
        for (int off = 8; off >= 1; off >>= 1) {
            float om = __shfl_xor(sm[r], off, 32);
            float oS = __shfl_xor(ss[r], off, 32);
            int   oa = __shfl_xor(sa[r], off, 32);
            comb(sm[r], ss[r], sa[r], om, oS, oa);
        }
    }

    // cross-wave reduction via LDS (ascending wave = ascending n range)
    __shared__ float wm[8][16];
    __shared__ float wS[8][16];
    __shared__ int   wa[8][16];
    if (l15 == 0) {
        #pragma unroll
        for (int r = 0; r < 8; ++r) {
            int row = half * 8 + r;
            wm[wv][row] = sm[r];
            wS[wv][row] = ss[r];
            wa[wv][row] = sa[r];
        }
    }
    __syncthreads();
    if (tid < 16) {
        float M = -__builtin_inff(), S = 0.0f;
        int   A = 0;
        #pragma unroll
        for (int w = 0; w < 8; ++w) comb(M, S, A, wm[w][tid], wS[w][tid], wa[w][tid]);
        int s   = stile * 16 + tid;
        int idx = s * NCHUNK + chunk;
        pm[idx] = M; ps[idx] = S; pa[idx] = A;
    }
}

// ---------------- combine chunk partials, emit one-hot value ----------------
__global__ __launch_bounds__(256)
void reduce_kernel(const float* __restrict__ pm, const float* __restrict__ ps,
                   const int* __restrict__ pa, int t,
                   float* __restrict__ out,
                   int* __restrict__ amax, float* __restrict__ yval) {
    int s = blockIdx.x * blockDim.x + threadIdx.x;
    if (s >= NS) return;
    float M = -__builtin_inff(), S = 0.0f;
    int   A = 0;
    for (int ch = 0; ch < NCHUNK; ++ch)
        comb(M, S, A, pm[s * NCHUNK + ch], ps[s * NCHUNK + ch], pa[s * NCHUNK + ch]);
    float p  = 1.0f / S;                 // softmax prob at argmax
    float yv = (1.0f - p) + p;           // straight-through forward value
    out[((long)s * RW + t) * N_VOCAB + A] = yv;
    amax[s] = A;
    yval[s] = yv;
}

// ---------------------------------------------------------------------------
extern "C" void kernel_launch(void* const* d_in, const int* in_sizes, int n_in,
                              void* d_out, int out_size, void* d_ws, size_t ws_size,
                              hipStream_t stream) {
    const float* z      = (const float*)d_in[0];
    const float* u      = (const float*)d_in[1];   // gumbel_u [RW,NS,N]
    const float* l1w    = (const float*)d_in[2];
    const float* l1b    = (const float*)d_in[3];
    const float* l2w    = (const float*)d_in[4];
    const float* l2b    = (const float*)d_in[5];
    const float* l3w    = (const float*)d_in[6];
    const float* l3b    = (const float*)d_in[7];
    const float* w_ih   = (const float*)d_in[8];
    const float* w_hh   = (const float*)d_in[9];
    const float* b_ih   = (const float*)d_in[10];
    const float* b_hh   = (const float*)d_in[11];
    const float* W_up   = (const float*)d_in[12];
    const float* b_up   = (const float*)d_in[13];
    const float* W_down = (const float*)d_in[14];
    float* out = (float*)d_out;

    // carve workspace (all 4-byte aligned)
    float* ws   = (float*)d_ws;
    float* h    = ws;                       ws += NS * H;
    float* c    = ws;                       ws += NS * H;
    float* yval = ws;                       ws += NS;
    float* pm   = ws;                       ws += NS * NCHUNK;
    float* ps   = ws;                       ws += NS * NCHUNK;
    int*   amax = (int*)ws;                 ws += NS;
    int*   pa   = (int*)ws;

    // 1) zero the full output (forward values are exactly 0 off-argmax)
    long n4 = (long)NS * RW * N_VOCAB / 4;
    zero_out_kernel<<<8192, 256, 0, stream>>>((float4*)out, n4);

    // 2) initial state h0/c0 from noise; x0 = 0 via (amax=0, yval=0)
    init_kernel<<<NS, 64, 0, stream>>>(z, l1w, l1b, l2w, l2b, l3w, l3b,
                                       h, c, amax, yval);

    // 3) 8 sequential steps: LSTM -> WMMA logits + gumbel + online softmax -> emit
    for (int t = 0; t < RW; ++t) {
        lstm_kernel<<<NS, 160, 0, stream>>>(w_ih, w_hh, b_ih, b_hh, W_down,
                                            amax, yval, h, c);
        logits_kernel<<<dim3(STILES, NCHUNK), 256, 0, stream>>>(
            h, W_up, b_up, u, t, pm, ps, pa);
        reduce_kernel<<<(NS + 255) / 256, 256, 0, stream>>>(
            pm, ps, pa, t, out, amax, yval);
    }
}